// NormalComponentReconstructionLoss_45131516346687
// MI455X (gfx1250) — compile-verified
//
#include <hip/hip_runtime.h>
#include <hip/hip_bf16.h>

typedef __attribute__((ext_vector_type(2))) float v2f;
typedef __attribute__((ext_vector_type(8))) float v8f;

#define NBLOCKS 1024   // fixed so partial count is a multiple of 64 for the WMMA reducer
#define NTHREADS 256   // 8 waves per block (wave32)

// ---------------------------------------------------------------------------
// Kernel 1: streaming evaluation of |r - H(H^T r)|^2, one partial per block.
// Bandwidth-bound: 48 B/point read, ~33 FLOP/point. Deterministic partition.
// ---------------------------------------------------------------------------
__global__ __launch_bounds__(NTHREADS) void ncrl_partial_kernel(
    const float* __restrict__ pts,   // [n,3]
    const float* __restrict__ rec,   // [n,3]
    const float* __restrict__ frm,   // [n,3,2]
    float* __restrict__ partial,     // [NBLOCKS]
    int n)
{
    const int tid    = blockIdx.x * blockDim.x + threadIdx.x;
    const int stride = gridDim.x * blockDim.x;

    float acc = 0.0f;
    for (int i = tid; i < n; i += stride) {
        // 12-byte records, lanes contiguous -> coalesced b96-class loads
        float p[3], q[3], h[6];
        __builtin_memcpy(p, pts + (size_t)3 * i, 12);
        __builtin_memcpy(q, rec + (size_t)3 * i, 12);
        // 24-byte record -> b128 + b64 class loads
        __builtin_memcpy(h, frm + (size_t)6 * i, 24);

        const float r0 = q[0] - p[0];
        const float r1 = q[1] - p[1];
        const float r2 = q[2] - p[2];

        // c = H^T r   (H row-major [i][j]: h[2*i+j])
        const float c0 = fmaf(h[4], r2, fmaf(h[2], r1, h[0] * r0));
        const float c1 = fmaf(h[5], r2, fmaf(h[3], r1, h[1] * r0));

        // y = H c ; d = r - y
        const float d0 = r0 - fmaf(h[1], c1, h[0] * c0);
        const float d1 = r1 - fmaf(h[3], c1, h[2] * c0);
        const float d2 = r2 - fmaf(h[5], c1, h[4] * c0);

        acc = fmaf(d0, d0, acc);
        acc = fmaf(d1, d1, acc);
        acc = fmaf(d2, d2, acc);
    }

    // Block reduction through LDS (ds ops + workgroup barrier)
    __shared__ float sm[NTHREADS];
    sm[threadIdx.x] = acc;
    __syncthreads();
    #pragma unroll
    for (int s = NTHREADS / 2; s > 0; s >>= 1) {
        if (threadIdx.x < s) sm[threadIdx.x] += sm[threadIdx.x + s];
        __syncthreads();
    }
    if (threadIdx.x == 0) partial[blockIdx.x] = sm[0];
}

// ---------------------------------------------------------------------------
// Kernel 2: one wave32 reduces NBLOCKS partials with V_WMMA_F32_16X16X4_F32.
// A = ones(16x4); B packs 64 partials/iteration (2 VGPRs x 32 lanes, bijective
// slot->(k,n) mapping), C accumulates. Row 0 of D holds the 16 column sums;
// their sum is the grand total regardless of the exact lane layout.
// EXEC is all-ones at the WMMA: blockDim==32, uniform control flow.
// ---------------------------------------------------------------------------
__global__ __launch_bounds__(32) void ncrl_reduce_wmma_kernel(
    const float* __restrict__ partial,  // [NBLOCKS]
    float* __restrict__ out,            // [1]
    float invN)
{
    const int lane = threadIdx.x;

    v2f a;
    a[0] = 1.0f;
    a[1] = 1.0f;
    v8f c = {};

    #pragma unroll 4
    for (int base = 0; base < NBLOCKS; base += 64) {
        v2f b;
        b[0] = partial[base + lane];
        b[1] = partial[base + 32 + lane];
        // 8 args: (neg_a, A, neg_b, B, c_mod, C, reuse_a, reuse_b)
        c = __builtin_amdgcn_wmma_f32_16x16x4_f32(
                /*neg_a=*/false, a, /*neg_b=*/false, b,
                /*c_mod=*/(short)0, c, /*reuse_a=*/false, /*reuse_b=*/false);
    }

    // D VGPR0: lanes 0..15 hold D[0][N=lane] = colsum_N. Sum those 16.
    __shared__ float red[32];
    red[lane] = c[0];
    __syncthreads();
    if (lane == 0) {
        float s = 0.0f;
        #pragma unroll
        for (int k = 0; k < 16; ++k) s += red[k];
        out[0] = s * invN;
    }
}

extern "C" void kernel_launch(void* const* d_in, const int* in_sizes, int n_in,
                              void* d_out, int out_size, void* d_ws, size_t ws_size,
                              hipStream_t stream) {
    const float* pts = (const float*)d_in[0];  // [N,3]
    const float* rec = (const float*)d_in[1];  // [N,3]
    const float* frm = (const float*)d_in[2];  // [N,3,2]
    float* out = (float*)d_out;

    const int n = in_sizes[0] / 3;             // N points
    float* partial = (float*)d_ws;             // NBLOCKS floats of scratch

    ncrl_partial_kernel<<<NBLOCKS, NTHREADS, 0, stream>>>(pts, rec, frm, partial, n);
    ncrl_reduce_wmma_kernel<<<1, 32, 0, stream>>>(partial, out, 1.0f / (float)n);
}